// CellLineCaps_57389353009433
// MI455X (gfx1250) — compile-verified
//
#include <hip/hip_runtime.h>

// CapsNet dynamic routing, fused for MI455X (gfx1250, wave32, WMMA).
// Roofline: recompute u_hat from x (9.4MB) + W (5.9MB, L2-resident) instead of
// materializing 189MB — ~75MB total traffic vs >1GB. All contractions on
// v_wmma_f32_16x16x4_f32 (fp32-exact vs reference). Dual accumulators break
// the D->C dependent chain; 8 waves/block split routes for occupancy.

#define R_ 1152
#define C_ 10
#define I_ 8
#define O_ 16
#define B_ 256
#define ITERS_ 3

typedef __attribute__((ext_vector_type(2))) float v2f;
typedef __attribute__((ext_vector_type(8))) float v8f;

__global__ void k_zero(float* __restrict__ p, int n) {
    int i = blockIdx.x * blockDim.x + threadIdx.x;
    if (i < n) p[i] = 0.0f;
}

// softmax over routes (axis 0) of b[R,C]; one block per capsule column
__global__ void k_softmax_routes(const float* __restrict__ b, float* __restrict__ c) {
    __shared__ float red[256];
    const int cap = blockIdx.x;
    const int t = threadIdx.x;
    float m = -1e30f;
    for (int r = t; r < R_; r += 256) m = fmaxf(m, b[r * C_ + cap]);
    red[t] = m; __syncthreads();
    for (int s = 128; s > 0; s >>= 1) { if (t < s) red[t] = fmaxf(red[t], red[t + s]); __syncthreads(); }
    m = red[0]; __syncthreads();
    float sum = 0.0f;
    for (int r = t; r < R_; r += 256) sum += __expf(b[r * C_ + cap] - m);
    red[t] = sum; __syncthreads();
    for (int s = 128; s > 0; s >>= 1) { if (t < s) red[t] += red[t + s]; __syncthreads(); }
    const float inv = 1.0f / red[0];
    for (int r = t; r < R_; r += 256) c[r * C_ + cap] = __expf(b[r * C_ + cap] - m) * inv;
}

// Fused s_j + squash: one block per (16-batch tile, capsule), 8 waves.
// s[b,o] = sum_r c[r,cap] * sum_i W[r,cap,o,i] * x[b,r,i]
// Routing weight folded into A operand -> pure WMMA accumulation chains.
__global__ __launch_bounds__(256)
void k_s_squash(const float* __restrict__ x, const float* __restrict__ W,
                const float* __restrict__ c, float* __restrict__ v,
                float* __restrict__ vout, int write_out) {
    const int btile = blockIdx.x;       // 0..15
    const int cap   = blockIdx.y;       // 0..9
    const int t     = threadIdx.x;      // 0..255 (8 waves)
    const int wave  = t >> 5;
    const int lane  = t & 31;
    const int hl    = lane >> 4;        // half-wave select
    const int l16   = lane & 15;

    __shared__ float partial[8][8][32];
    __shared__ float stile[16][16];
    __shared__ float norm2[16];

    v8f acc0 = {};                      // K = 0..3 leg
    v8f acc1 = {};                      // K = 4..7 leg (independent chain)
    const int bA = btile * 16 + l16;    // A row M = batch index
    const int k0 = 2 * hl;              // K base within the 4-wide step

    for (int r = wave; r < R_; r += 8) {
        const float cr = c[r * C_ + cap];
        const float* xp = x + (size_t)bA * (R_ * I_) + r * I_;
        const float* wp = W + (size_t)r * (C_ * O_ * I_) + cap * (O_ * I_) + l16 * I_;
        // prefetch next route's W slab for this wave (global_prefetch_b8)
        __builtin_prefetch(wp + 8 * (C_ * O_ * I_), 0, 3);

        // K = 0..3 (i = k)
        v2f a0; a0.x = cr * xp[k0 + 0]; a0.y = cr * xp[k0 + 1];
        v2f b0; b0.x = wp[k0 + 0];      b0.y = wp[k0 + 1];   // B[k=i][n=o] = W[r,cap,o,i]
        acc0 = __builtin_amdgcn_wmma_f32_16x16x4_f32(false, a0, false, b0, (short)0, acc0, false, false);
        // K = 4..7
        v2f a1; a1.x = cr * xp[k0 + 4]; a1.y = cr * xp[k0 + 5];
        v2f b1; b1.x = wp[k0 + 4];      b1.y = wp[k0 + 5];
        acc1 = __builtin_amdgcn_wmma_f32_16x16x4_f32(false, a1, false, b1, (short)0, acc1, false, false);
    }
    const v8f acc = acc0 + acc1;

    // cross-wave reduce C/D tiles (layout: M = vgpr + 8*hl, N = l16)
#pragma unroll
    for (int vv = 0; vv < 8; ++vv) partial[wave][vv][lane] = acc[vv];
    __syncthreads();

    {   // each of 256 threads combines one (b,o) element across the 8 waves
        const int bl = t >> 4, o = t & 15;
        const int vv = bl & 7;
        const int ln = ((bl >> 3) << 4) | o;
        float s = 0.0f;
#pragma unroll
        for (int w = 0; w < 8; ++w) s += partial[w][vv][ln];
        stile[bl][o] = s;
    }
    __syncthreads();

    if (t < 16) {
        float n2 = 0.0f;
#pragma unroll
        for (int o = 0; o < 16; ++o) { const float s = stile[t][o]; n2 += s * s; }
        norm2[t] = n2;
    }
    __syncthreads();

    {
        const int bl = t >> 4, o = t & 15;
        const float n2 = norm2[bl];
        const float s  = stile[bl][o];
        const float val = (n2 / (1.0f + n2)) * s * rsqrtf(n2 + 1e-9f);
        const int idx = ((btile * 16 + bl) * C_ + cap) * O_ + o;
        v[idx] = val;
        if (write_out) vout[idx] = val;
    }
}

// G[co, ri] = sum_b v[b, co] * x[b, ri]   -> [160 x 9216] GEMM, K = 256
__global__ __launch_bounds__(32)
void k_grad_gemm(const float* __restrict__ v, const float* __restrict__ x,
                 float* __restrict__ G) {
    const int ntile = blockIdx.x;   // 0..575
    const int mtile = blockIdx.y;   // 0..9
    const int lane  = threadIdx.x;
    const int hl    = lane >> 4;
    const int l16   = lane & 15;
    const int co = mtile * 16 + l16;    // A row (cap*16+o)
    const int ri = ntile * 16 + l16;    // B col (r*8+i)

    v8f acc0 = {};
    v8f acc1 = {};
    for (int ks = 0; ks < B_; ks += 8) {
        const int ka = ks + 2 * hl;         // K = batch, first 4-wide step
        const int kb = ka + 4;              // second step -> independent chain
        v2f a0; a0.x = v[(ka + 0) * (C_ * O_) + co];
                a0.y = v[(ka + 1) * (C_ * O_) + co];
        v2f b0; b0.x = x[(size_t)(ka + 0) * (R_ * I_) + ri];
                b0.y = x[(size_t)(ka + 1) * (R_ * I_) + ri];
        acc0 = __builtin_amdgcn_wmma_f32_16x16x4_f32(false, a0, false, b0, (short)0, acc0, false, false);
        v2f a1; a1.x = v[(kb + 0) * (C_ * O_) + co];
                a1.y = v[(kb + 1) * (C_ * O_) + co];
        v2f b1; b1.x = x[(size_t)(kb + 0) * (R_ * I_) + ri];
                b1.y = x[(size_t)(kb + 1) * (R_ * I_) + ri];
        acc1 = __builtin_amdgcn_wmma_f32_16x16x4_f32(false, a1, false, b1, (short)0, acc1, false, false);
    }
    const v8f acc = acc0 + acc1;
#pragma unroll
    for (int vv = 0; vv < 8; ++vv) {
        const int row = mtile * 16 + vv + 8 * hl;
        G[(size_t)row * (R_ * I_) + ri] = acc[vv];
    }
}

// b[r,c] += (1/B) * sum_{o,i} W[r,c,o,i] * G[c*16+o, r*8+i]
__global__ void k_b_update(const float* __restrict__ W, const float* __restrict__ G,
                           float* __restrict__ b) {
    const int idx = blockIdx.x * blockDim.x + threadIdx.x;
    if (idx >= R_ * C_) return;
    const int r = idx / C_, cap = idx % C_;
    const float* wp = W + (size_t)r * (C_ * O_ * I_) + cap * (O_ * I_);
    float acc = 0.0f;
#pragma unroll
    for (int o = 0; o < O_; ++o) {
        const float* gp = G + (size_t)(cap * O_ + o) * (R_ * I_) + r * I_;
#pragma unroll
        for (int i = 0; i < I_; ++i) acc += wp[o * I_ + i] * gp[i];
    }
    b[idx] += acc * (1.0f / B_);
}

// c_out[b, r, c] = c[r, c] broadcast over batch
__global__ void k_cout(const float* __restrict__ c, float* __restrict__ out) {
    const int i = blockIdx.x * blockDim.x + threadIdx.x;
    if (i < B_ * R_ * C_) out[i] = c[i % (R_ * C_)];
}

extern "C" void kernel_launch(void* const* d_in, const int* in_sizes, int n_in,
                              void* d_out, int out_size, void* d_ws, size_t ws_size,
                              hipStream_t stream) {
    const float* x = (const float*)d_in[0];   // [B, R, I]
    const float* W = (const float*)d_in[1];   // [1, R, C, O, I]
    float* out = (float*)d_out;               // [B*C*O] then [B*R*C]
    float* ws  = (float*)d_ws;

    float* c = ws;                      // R*C   = 11520
    float* b = ws + R_ * C_;            // R*C   = 11520
    float* v = b + R_ * C_;             // B*C*O = 40960
    float* G = v + B_ * C_ * O_;        // 160*9216 = 1474560  (~6 MB total)

    k_zero<<<(R_ * C_ + 255) / 256, 256, 0, stream>>>(b, R_ * C_);

    for (int it = 0; it < ITERS_; ++it) {
        k_softmax_routes<<<C_, 256, 0, stream>>>(b, c);
        dim3 gs(B_ / 16, C_);
        k_s_squash<<<gs, 256, 0, stream>>>(x, W, c, v, out, (it == ITERS_ - 1) ? 1 : 0);
        if (it < ITERS_ - 1) {
            dim3 gg((R_ * I_) / 16, (C_ * O_) / 16);
            k_grad_gemm<<<gg, 32, 0, stream>>>(v, x, G);
            k_b_update<<<(R_ * C_ + 255) / 256, 256, 0, stream>>>(W, G, b);
        }
    }
    k_cout<<<(B_ * R_ * C_ + 255) / 256, 256, 0, stream>>>(c, out + B_ * C_ * O_);
}